// GraphAttentionHeadModule_1666447311044
// MI455X (gfx1250) — compile-verified
//
#include <hip/hip_runtime.h>
#include <math.h>

typedef __attribute__((ext_vector_type(2))) float v2f;
typedef __attribute__((ext_vector_type(8))) float v8f;
typedef __attribute__((address_space(3))) char lds_char;

#define N_NODES 8192
#define D_IN    512
#define D_OUT   256
#define NEG_INF_V (-9e15f)
#define LRELU_A 0.2f

#define ROWS 32   // attention rows per workgroup (K3)
#define JT   32   // column tile (K3)
#define JPT  4    // columns per softmax thread
#define PSTR 36   // p_s row stride (floats): 36 mod 64 -> conflict-free b64 A loads
#define HSTR 264  // h_s row stride (floats): 8 mod 64 -> disjoint half-wave bank windows

// 32-bit LDS byte offset of a shared-memory pointer (for async-to-LDS dst)
static __device__ __forceinline__ unsigned lds_off(void* p) {
  return (unsigned)(unsigned long long)(lds_char*)p;
}

// ---------------------------------------------------------------------------
// K1: H = X @ W   (8192x512 @ 512x256), f32 WMMA 16x16x4, one 16x16 tile/wave
// ---------------------------------------------------------------------------
__global__ __launch_bounds__(256) void k_gemm_xw(const float* __restrict__ X,
                                                 const float* __restrict__ W,
                                                 float* __restrict__ H) {
  const int tid  = threadIdx.x;
  const int lane = tid & 31;
  const int wave = tid >> 5;
  const int tile = blockIdx.x * 8 + wave;   // (N/16)*(D_OUT/16) = 8192 tiles
  const int tn   = tile & 15;
  const int tm   = tile >> 4;
  const int l15  = lane & 15;
  const int koff = (lane >> 4) * 2;

  v8f acc = {};
  const float* xrow = X + (size_t)(tm * 16 + l15) * D_IN;
  const float* wcol = W + tn * 16 + l15;

  for (int k0 = 0; k0 < D_IN; k0 += 4) {
    float2 av = *(const float2*)(xrow + k0 + koff);
    v2f A; A.x = av.x; A.y = av.y;
    v2f B;
    B.x = wcol[(size_t)(k0 + koff) * D_OUT];
    B.y = wcol[(size_t)(k0 + koff + 1) * D_OUT];
    acc = __builtin_amdgcn_wmma_f32_16x16x4_f32(false, A, false, B,
                                                (short)0, acc, false, false);
  }

  const int mbase = tm * 16 + (lane >> 4) * 8;
  float* hout = H + (size_t)mbase * D_OUT + tn * 16 + l15;
#pragma unroll
  for (int r = 0; r < 8; ++r)
    hout[(size_t)r * D_OUT] = acc[r];
}

// ---------------------------------------------------------------------------
// K2: mu[i] = h[i,:].w_mu ; xi[i] = h[i,:].w_xi   (one row per wave)
// ---------------------------------------------------------------------------
__global__ __launch_bounds__(256) void k_mu_xi(const float* __restrict__ H,
                                               const float* __restrict__ wmu,
                                               const float* __restrict__ wxi,
                                               float* __restrict__ mu,
                                               float* __restrict__ xi) {
  const int lane = threadIdx.x & 31;
  const int wave = threadIdx.x >> 5;
  const int row  = blockIdx.x * 8 + wave;
  const float* hrow = H + (size_t)row * D_OUT;
  float sm = 0.f, sx = 0.f;
  for (int c = lane; c < D_OUT; c += 32) {
    float hv = hrow[c];
    sm += hv * wmu[c];
    sx += hv * wxi[c];
  }
#pragma unroll
  for (int m = 16; m >= 1; m >>= 1) {
    sm += __shfl_xor(sm, m, 32);
    sx += __shfl_xor(sx, m, 32);
  }
  if (lane == 0) { mu[row] = sm; xi[row] = sx; }
}

// ---------------------------------------------------------------------------
// K3: fused masked-attention (flash style). 8 waves in a 2x4 grid over a
// 32x256 f32 accumulator. H tile staged in LDS via async global->LDS b128
// copies (ASYNCcnt), overlapped with the softmax VALU phase; the hot K-loop
// is pure ds_load + v_wmma_f32_16x16x4_f32.
// ---------------------------------------------------------------------------
__global__ __launch_bounds__(256) void k_flash(const float* __restrict__ H,
                                               const int*   __restrict__ adj,
                                               const float* __restrict__ mu,
                                               const float* __restrict__ xi,
                                               float* __restrict__ out) {
  __shared__ float p_s[ROWS][PSTR];   // softmax weights tile
  __shared__ float h_s[JT][HSTR];     // staged H tile (JT x 256, padded)
  __shared__ float c_s[ROWS];
  __shared__ float l_s[ROWS];

  const int tid  = threadIdx.x;
  const int lane = tid & 31;
  const int wave = tid >> 5;
  const int wrow = wave >> 2;            // 0..1  -> 16-row group
  const int wcol = wave & 3;             // 0..3  -> 64-col group
  const int l15  = lane & 15;
  const int half = lane >> 4;            // 0/1
  const int koff = half * 2;

  const int rowbase = blockIdx.x * ROWS;

  // softmax mapping: 8 threads per row, 4 columns per thread
  const int r  = tid >> 3;               // 0..31
  const int jj = (tid & 7) * JPT;        // 0,4,...,28
  const float mu_r = mu[rowbase + r];
  const int* arow_g = adj + (size_t)(rowbase + r) * N_NODES;

  // cooperative h-tile copy mapping: 8 x b128 per thread
  const int crow0 = tid >> 6;            // 0..3 (row step 4 per chunk)
  const int cc4   = tid & 63;            // float4 column
  const unsigned lds_dst0 = lds_off(&h_s[crow0][cc4 * 4]);
  const float* g_src0 = H + (size_t)crow0 * D_OUT + cc4 * 4;

  float m_run = -INFINITY;
  float l_run = 0.f;
  v8f acc[4] = {{}, {}, {}, {}};

  const int a_row   = wrow * 16 + l15;        // p_s row for A fragments
  const int c_row_l = wrow * 16 + half * 8;   // base M row of this lane's acc

  for (int jb = 0; jb < N_NODES; jb += JT) {
    __syncthreads();   // previous tile's p_s / h_s fully consumed

    // ---- issue async copy of H tile (JT x 256 f32) into LDS ----
    {
      const float* gp = g_src0 + (size_t)jb * D_OUT;
#pragma unroll
      for (int i = 0; i < 8; ++i) {
        unsigned long long ga = (unsigned long long)(gp + (size_t)i * 4 * D_OUT);
        unsigned la = lds_dst0 + (unsigned)(i * 4 * HSTR * 4);
        asm volatile("global_load_async_to_lds_b128 %0, %1, off"
                     :: "v"(la), "v"(ga) : "memory");
      }
    }

    if (jb + JT < N_NODES)
      __builtin_prefetch(arow_g + jb + JT + jj, 0, 0);  // next adjacency tile

    // ---- scores for 4 columns (overlaps with async copy) ----
    int4   a0 = *(const int4*)  (arow_g + jb + jj);
    float4 x0 = *(const float4*)(xi + jb + jj);
    float e[4];
    {
      const int   am[4] = {a0.x, a0.y, a0.z, a0.w};
      const float xv[4] = {x0.x, x0.y, x0.z, x0.w};
#pragma unroll
      for (int k = 0; k < 4; ++k) {
        float ev = mu_r + xv[k];
        ev = ev >= 0.f ? ev : LRELU_A * ev;   // LeakyReLU
        e[k] = (am[k] > 0) ? ev : NEG_INF_V;  // mask non-edges
      }
    }

    // ---- online softmax (8-lane row groups, same wave) ----
    float tmax = fmaxf(fmaxf(e[0], e[1]), fmaxf(e[2], e[3]));
    tmax = fmaxf(tmax, __shfl_xor(tmax, 1, 32));
    tmax = fmaxf(tmax, __shfl_xor(tmax, 2, 32));
    tmax = fmaxf(tmax, __shfl_xor(tmax, 4, 32));

    const float mnew = fmaxf(m_run, tmax);
    const float cf   = __expf(m_run - mnew);
    float tsum = 0.f;
#pragma unroll
    for (int k = 0; k < 4; ++k) {
      float p = __expf(e[k] - mnew);
      p_s[r][jj + k] = p;
      tsum += p;
    }
    tsum += __shfl_xor(tsum, 1, 32);
    tsum += __shfl_xor(tsum, 2, 32);
    tsum += __shfl_xor(tsum, 4, 32);
    l_run = l_run * cf + tsum;
    m_run = mnew;
    if ((tid & 7) == 0) c_s[r] = cf;

    // ---- wait for this wave's async tile writes, then block-wide sync ----
    asm volatile("s_wait_asynccnt 0" ::: "memory");
    __syncthreads();

    // ---- rescale accumulators by per-row correction ----
    float cv[8];
#pragma unroll
    for (int q = 0; q < 8; ++q) cv[q] = c_s[c_row_l + q];
#pragma unroll
    for (int t = 0; t < 4; ++t)
#pragma unroll
      for (int q = 0; q < 8; ++q) acc[t][q] *= cv[q];

    // ---- acc += P(32xJT) @ Htile(JT x 256), all from LDS ----
#pragma unroll
    for (int kt = 0; kt < JT / 4; ++kt) {
      const int kk = kt * 4 + koff;
      float2 av = *(const float2*)(&p_s[a_row][kk]);   // ds_load_b64
      v2f A; A.x = av.x; A.y = av.y;
      const float* hb = &h_s[kk][wcol * 64 + l15];
#pragma unroll
      for (int t = 0; t < 4; ++t) {
        v2f B;
        B.x = hb[t * 16];          // ds_load_b32, conflict-free layout
        B.y = hb[HSTR + t * 16];
        acc[t] = __builtin_amdgcn_wmma_f32_16x16x4_f32(false, A, false, B,
                                                       (short)0, acc[t],
                                                       false, false);
      }
    }
  }

  // ---- epilogue: divide by l, ELU, store ----
  if ((tid & 7) == 0) l_s[r] = l_run;
  __syncthreads();

  float lv[8];
#pragma unroll
  for (int q = 0; q < 8; ++q) lv[q] = l_s[c_row_l + q];
#pragma unroll
  for (int t = 0; t < 4; ++t) {
#pragma unroll
    for (int q = 0; q < 8; ++q) {
      float v = acc[t][q] / lv[q];
      v = v > 0.f ? v : expm1f(v);   // ELU (alpha=1)
      const int row_g = rowbase + c_row_l + q;
      const int col_g = wcol * 64 + t * 16 + l15;
      out[(size_t)row_g * D_OUT + col_g] = v;
    }
  }
}

// ---------------------------------------------------------------------------
extern "C" void kernel_launch(void* const* d_in, const int* in_sizes, int n_in,
                              void* d_out, int out_size, void* d_ws, size_t ws_size,
                              hipStream_t stream) {
  const float* X   = (const float*)d_in[0];   // features     [8192,512]
  const int*   adj = (const int*)  d_in[1];   // adjacency    [8192,8192]
  const float* W   = (const float*)d_in[2];   // W_phi        [512,256]
  const float* wmu = (const float*)d_in[3];   // w_mu         [256]
  const float* wxi = (const float*)d_in[4];   // w_xi         [256]
  float* out = (float*)d_out;                 // [8192,256]

  float* H  = (float*)d_ws;                   // 8192*256 f32 = 8 MB
  float* mu = H + (size_t)N_NODES * D_OUT;    // 8192 f32
  float* xi = mu + N_NODES;                   // 8192 f32

  k_gemm_xw<<<(N_NODES / 16) * (D_OUT / 16) / 8, 256, 0, stream>>>(X, W, H);
  k_mu_xi<<<N_NODES / 8, 256, 0, stream>>>(H, wmu, wxi, mu, xi);
  k_flash<<<N_NODES / ROWS, 256, 0, stream>>>(H, adj, mu, xi, out);
}